// L2Net_26147760898164
// MI455X (gfx1250) — compile-verified
//
#include <hip/hip_runtime.h>
#include <hip/hip_bf16.h>
#include <math.h>

typedef float v2f __attribute__((ext_vector_type(2)));
typedef float v8f __attribute__((ext_vector_type(8)));

#define TAU_M 10.0f

__device__ __forceinline__ v8f zero8() {
    v8f z;
#pragma unroll
    for (int i = 0; i < 8; ++i) z[i] = 0.0f;
    return z;
}

// One wave (32 lanes) per block; block b owns rows nc = b*16 .. b*16+15
// (all same n, consecutive c). Full pipeline fused over tau = 0..95.
__global__ __launch_bounds__(32) void snn_main(
    const float* __restrict__ x,      // (64,128,2,128)
    const float* __restrict__ Wjeff,  // (33,2)
    const float* __restrict__ Wamp,   // (33,33)
    const float* __restrict__ wsyn1,  // (1,)
    const float* __restrict__ Wlin,   // (33,1)
    const float* __restrict__ wsyn2,  // (1,)
    const float* __restrict__ Wout,   // (1,1)
    float* __restrict__ partial)      // (64, 1024) block partial c-sums
{
    __shared__ float s1lds[36 * 16];   // K-padded spike tile  [k][row]
    __shared__ float wamplds[36 * 48]; // padded W_amp         [k][o]

    const int lane = threadIdx.x;
    const int row  = lane & 15;   // M row within tile / o column within tile
    const int lh   = lane >> 4;   // lane half (WMMA K interleave)
    const int nc0  = blockIdx.x * 16;
    const int n    = nc0 >> 7;
    const int c    = (nc0 & 127) + row;

    // ---- stage weights into LDS (zero-padded) ----
    for (int idx = lane; idx < 36 * 48; idx += 32) {
        int k = idx / 48, o = idx % 48;
        wamplds[idx] = (k < 33 && o < 33) ? Wamp[k * 33 + o] : 0.0f;
    }
    for (int idx = lane; idx < 36 * 16; idx += 32) s1lds[idx] = 0.0f;
    __syncthreads();

    // ---- per-lane constants ----
    float wj0[17], wj1[17];
#pragma unroll
    for (int jj = 0; jj < 17; ++jj) {
        int j = lh * 17 + jj;
        bool ok = (j < 33) && (lh == 0 || jj < 16);
        wj0[jj] = ok ? Wjeff[j * 2 + 0] : 0.0f;
        wj1[jj] = ok ? Wjeff[j * 2 + 1] : 0.0f;
    }
    const float dec1 = 1.0f - 1.0f / (1.0f + expf(-wsyn1[0])); // 1 - sigmoid
    const float dec2 = 1.0f - 1.0f / (1.0f + expf(-wsyn2[0]));
    const float wout = Wout[0];
    float wlin[3];
#pragma unroll
    for (int ot = 0; ot < 3; ++ot) {
        int o = ot * 16 + row;
        wlin[ot] = (o < 33) ? Wlin[o] : 0.0f;
    }

    // ---- state ----
    float v1[17];                       // layer-1 LIF (per lane: its j's)
#pragma unroll
    for (int i = 0; i < 17; ++i) v1[i] = 0.0f;
    float v2[24], ys[24];               // layer-2 LIF + synapse filter, D-layout
#pragma unroll
    for (int i = 0; i < 24; ++i) { v2[i] = 0.0f; ys[i] = 0.0f; }
    float v3[8], fs[8];                 // layer-3 LIF + output filter (per row)
#pragma unroll
    for (int i = 0; i < 8; ++i) { v3[i] = 0.0f; fs[i] = 0.0f; }

    // ---- fused pipelined time loop ----
    for (int tau = 0; tau < 96; ++tau) {
        float s2cur[24];
        if (tau < 80) {
            // stage 1: delay taps + LIF -> spikes into LDS (A tile)
#pragma unroll
            for (int jj = 0; jj < 17; ++jj) {
                int j = lh * 17 + jj;
                if (lh == 0 || jj < 16) {
                    int tl = tau - j;
                    int tr = tau - 32 + j;
                    float xl = (tl >= 0 && tl < 64)
                                   ? x[((tl * 128 + n) * 2 + 0) * 128 + c] : 0.0f;
                    float xr = (tr >= 0 && tr < 64)
                                   ? x[((tr * 128 + n) * 2 + 1) * 128 + c] : 0.0f;
                    float u = wj0[jj] * xl + wj1[jj] * xr;
                    float h = v1[jj] + (u - v1[jj]) / TAU_M;
                    float s = (h >= 1.0f) ? 1.0f : 0.0f;
                    v1[jj] = h * (1.0f - s);
                    s1lds[j * 16 + row] = s;
                }
            }
            __syncthreads();

            // stage 1b: z = s1 @ W_amp via f32 WMMA, 16x48 tile, K=36
            v8f acc0 = zero8(), acc1 = zero8(), acc2 = zero8();
            const int o0 = row, o1 = 16 + row, o2 = 32 + row;
#pragma unroll
            for (int ks = 0; ks < 9; ++ks) {
                const int kb = ks * 4 + lh * 2; // K interleave across lane halves
                v2f a, b;
                a.x = s1lds[kb * 16 + row];
                a.y = s1lds[(kb + 1) * 16 + row];
                b.x = wamplds[kb * 48 + o0];
                b.y = wamplds[(kb + 1) * 48 + o0];
                acc0 = __builtin_amdgcn_wmma_f32_16x16x4_f32(
                    false, a, false, b, (short)0, acc0, false, false);
                b.x = wamplds[kb * 48 + o1];
                b.y = wamplds[(kb + 1) * 48 + o1];
                acc1 = __builtin_amdgcn_wmma_f32_16x16x4_f32(
                    false, a, false, b, (short)0, acc1, false, false);
                b.x = wamplds[kb * 48 + o2];
                b.y = wamplds[(kb + 1) * 48 + o2];
                acc2 = __builtin_amdgcn_wmma_f32_16x16x4_f32(
                    false, a, false, b, (short)0, acc2, false, false);
            }
            __syncthreads(); // protect s1lds for next iteration's stores

            // stage 2: LIF on z (elementwise in D-layout registers)
#pragma unroll
            for (int k = 0; k < 8; ++k) {
                float z0 = acc0[k], z1 = acc1[k], z2 = acc2[k];
                float h0 = v2[0 * 8 + k] + (z0 - v2[0 * 8 + k]) / TAU_M;
                float s0 = (h0 >= 1.0f) ? 1.0f : 0.0f;
                v2[0 * 8 + k] = h0 * (1.0f - s0); s2cur[0 * 8 + k] = s0;
                float h1 = v2[1 * 8 + k] + (z1 - v2[1 * 8 + k]) / TAU_M;
                float s1v = (h1 >= 1.0f) ? 1.0f : 0.0f;
                v2[1 * 8 + k] = h1 * (1.0f - s1v); s2cur[1 * 8 + k] = s1v;
                float h2 = v2[2 * 8 + k] + (z2 - v2[2 * 8 + k]) / TAU_M;
                float s2v = (h2 >= 1.0f) ? 1.0f : 0.0f;
                v2[2 * 8 + k] = h2 * (1.0f - s2v); s2cur[2 * 8 + k] = s2v;
            }
        } else {
#pragma unroll
            for (int i = 0; i < 24; ++i) s2cur[i] = 0.0f; // time_pad region
        }

        // stage 3 (t'' = tau-16): synapse filter, @W_lin, LIF
        if (tau >= 16) {
#pragma unroll
            for (int i = 0; i < 24; ++i) ys[i] = ys[i] * dec1 + s2cur[i];
#pragma unroll
            for (int k = 0; k < 8; ++k) {
                float p = ys[0 * 8 + k] * wlin[0] + ys[1 * 8 + k] * wlin[1] +
                          ys[2 * 8 + k] * wlin[2];
                // reduce over 16 columns within each lane half
                p += __shfl_xor(p, 1, 32);
                p += __shfl_xor(p, 2, 32);
                p += __shfl_xor(p, 4, 32);
                p += __shfl_xor(p, 8, 32);
                // y(row = k + 8*lh) now uniform within each 16-lane group
                float h = v3[k] + (p - v3[k]) / TAU_M;
                float s = (h >= 1.0f) ? 1.0f : 0.0f;
                v3[k] = h * (1.0f - s);
                // stage 4 (t3 = tau-32): output synapse filter
                if (tau >= 32) fs[k] = fs[k] * dec2 + s;
            }
            if (tau >= 32) {
                float tot = fs[0] + fs[1] + fs[2] + fs[3] +
                            fs[4] + fs[5] + fs[6] + fs[7]; // 8 rows of this half
                tot += __shfl_xor(tot, 16, 32);            // all 16 rows
                if (lane == 0)
                    partial[(tau - 32) * 1024 + blockIdx.x] = tot * wout;
            }
        }
    }
}

// Deterministic c-sum (8 block partials per n) + final leaky integrator.
__global__ __launch_bounds__(128) void snn_finalize(
    const float* __restrict__ partial, float* __restrict__ out)
{
    const int n = threadIdx.x;
    float v = 0.0f;
    for (int t = 0; t < 64; ++t) {
        float s = 0.0f;
#pragma unroll
        for (int b = 0; b < 8; ++b) s += partial[t * 1024 + n * 8 + b];
        v = v + (s - v) / TAU_M;
        out[t * 128 + n] = v;
    }
}

extern "C" void kernel_launch(void* const* d_in, const int* in_sizes, int n_in,
                              void* d_out, int out_size, void* d_ws, size_t ws_size,
                              hipStream_t stream) {
    const float* x     = (const float*)d_in[0]; // (64,128,2,128)
    const float* Wjeff = (const float*)d_in[1]; // (33,2)
    const float* Wamp  = (const float*)d_in[2]; // (33,33)
    const float* wsyn1 = (const float*)d_in[3]; // (1,)
    const float* Wlin  = (const float*)d_in[4]; // (33,1)
    const float* wsyn2 = (const float*)d_in[5]; // (1,)
    const float* Wout  = (const float*)d_in[6]; // (1,1)
    float* partial = (float*)d_ws;              // 64*1024 floats = 256 KB
    float* out     = (float*)d_out;             // (64,128,1)

    snn_main<<<1024, 32, 0, stream>>>(x, Wjeff, Wamp, wsyn1, Wlin, wsyn2, Wout,
                                      partial);
    snn_finalize<<<1, 128, 0, stream>>>(partial, out);
}